// LogSumExpWirelength_24816321037009
// MI455X (gfx1250) — compile-verified
//
#include <hip/hip_runtime.h>
#include <hip/hip_bf16.h>
#include <stdint.h>

// ---------------------------------------------------------------------------
// Weighted log-sum-exp wirelength (DREAMPlace-style), forward only.
//   WL = g * sum_n w_n * [ lse(x/g) + lse(-x/g) + lse(y/g) + lse(-y/g) ]
// Segmented max + segmented sum over 10M pins -> 2M nets.
//
// MI455X strategy:
//  * Per-net accumulators (4xN u32 maxes + 4xN f32 sums = 64 MB) stay resident
//    in the 192 MB L2; all 80M atomics resolve at L2 atomic throughput.
//  * Streaming pin data (240 MB over two passes) is read non-temporally
//    (TH=NT) so it never evicts the accumulator working set.
//  * 2 pins/thread with b64 vector loads; b128 loads of the interleaved
//    per-net quads.
//  * Hardware atomics: global_atomic_max_u32 (monotone-uint float encoding)
//    and global_atomic_add_f32 via unsafeAtomicAdd (avoids the CAS-loop
//    fallback clang emits for plain float atomicAdd).
//  * No WMMA: random scatter-reduce has no dense tile structure.
// ---------------------------------------------------------------------------

#define THREADS 256

typedef float    v2f __attribute__((ext_vector_type(2)));
typedef int      v2i __attribute__((ext_vector_type(2)));
typedef float    v4f __attribute__((ext_vector_type(4)));
typedef unsigned v4u __attribute__((ext_vector_type(4)));

// Hardware f32 atomic add (global_atomic_add_f32), not a CAS loop.
__device__ __forceinline__ void atomic_add_f32_hw(float* p, float v) {
#if defined(__HIP_DEVICE_COMPILE__)
    unsafeAtomicAdd(p, v);
#else
    atomicAdd(p, v);
#endif
}

// Monotone orderable encoding of f32 -> u32 (integer atomicMax == float max)
__device__ __forceinline__ unsigned enc_f32(float f) {
    unsigned u = __float_as_uint(f);
    return (u & 0x80000000u) ? ~u : (u | 0x80000000u);
}
__device__ __forceinline__ float dec_f32(unsigned u) {
    unsigned v = (u & 0x80000000u) ? (u & 0x7FFFFFFFu) : ~u;
    return __uint_as_float(v);
}
// enc_f32(-inf): bits(-inf)=0xFF800000 -> ~ -> 0x007FFFFF
#define ENC_NEG_INF 0x007FFFFFu

// ---------------------------------------------------------------------------
// Kernel 1: init accumulators (b128 stores) + zero output scalar
// ---------------------------------------------------------------------------
__global__ void k_init(v4u* __restrict__ enc_max4,
                       v4f* __restrict__ sums4,
                       float* __restrict__ out,
                       int N) {
    int i = blockIdx.x * blockDim.x + threadIdx.x;
    if (i == 0) out[0] = 0.0f;
    v4u m = {ENC_NEG_INF, ENC_NEG_INF, ENC_NEG_INF, ENC_NEG_INF};
    v4f z = {0.0f, 0.0f, 0.0f, 0.0f};
    for (; i < N; i += gridDim.x * blockDim.x) {
        enc_max4[i] = m;
        sums4[i]    = z;
    }
}

// ---------------------------------------------------------------------------
// Kernel 2: per-net max of (sx, -sx, sy, -sy) via integer atomic max.
// 2 pins per thread, b64 non-temporal streaming loads.
// ---------------------------------------------------------------------------
__global__ void k_max(const float* __restrict__ pos,
                      const int* __restrict__ p2n,
                      const float* __restrict__ gamma,
                      unsigned* __restrict__ enc_max,
                      int P) {
    int t  = blockIdx.x * blockDim.x + threadIdx.x;
    int p0 = t * 2;
    if (p0 >= P) return;
    float inv_g = 1.0f / gamma[0];

    if (((P & 1) == 0) && (p0 + 1 < P)) {
        v2f x = __builtin_nontemporal_load((const v2f*)pos + t);
        v2f y = __builtin_nontemporal_load((const v2f*)(pos + P) + t);
        v2i n = __builtin_nontemporal_load((const v2i*)p2n + t);
#pragma unroll
        for (int k = 0; k < 2; ++k) {
            float sx = x[k] * inv_g;
            float sy = y[k] * inv_g;
            unsigned* b = enc_max + 4u * (unsigned)n[k];
            atomicMax(b + 0, enc_f32( sx));
            atomicMax(b + 1, enc_f32(-sx));
            atomicMax(b + 2, enc_f32( sy));
            atomicMax(b + 3, enc_f32(-sy));
        }
    } else {
        for (int p = p0; p < P && p < p0 + 2; ++p) {
            float sx = pos[p] * inv_g;
            float sy = pos[P + p] * inv_g;
            unsigned* b = enc_max + 4u * (unsigned)p2n[p];
            atomicMax(b + 0, enc_f32( sx));
            atomicMax(b + 1, enc_f32(-sx));
            atomicMax(b + 2, enc_f32( sy));
            atomicMax(b + 3, enc_f32(-sy));
        }
    }
}

// ---------------------------------------------------------------------------
// Kernel 3: per-net sum of exp(s - max) via hardware f32 atomic add.
// Max quads re-read as regular (temporal) b128 loads: reused ~5x from L2.
// ---------------------------------------------------------------------------
__global__ void k_sum(const float* __restrict__ pos,
                      const int* __restrict__ p2n,
                      const float* __restrict__ gamma,
                      const unsigned* __restrict__ enc_max,
                      float* __restrict__ sums,
                      int P) {
    int t  = blockIdx.x * blockDim.x + threadIdx.x;
    int p0 = t * 2;
    if (p0 >= P) return;
    float inv_g = 1.0f / gamma[0];

    if (((P & 1) == 0) && (p0 + 1 < P)) {
        v2f x = __builtin_nontemporal_load((const v2f*)pos + t);
        v2f y = __builtin_nontemporal_load((const v2f*)(pos + P) + t);
        v2i n = __builtin_nontemporal_load((const v2i*)p2n + t);
#pragma unroll
        for (int k = 0; k < 2; ++k) {
            float sx = x[k] * inv_g;
            float sy = y[k] * inv_g;
            v4u m = *((const v4u*)enc_max + (unsigned)n[k]);   // b128, temporal
            float* sb = sums + 4u * (unsigned)n[k];
            atomic_add_f32_hw(sb + 0, __expf( sx - dec_f32(m.x)));
            atomic_add_f32_hw(sb + 1, __expf(-sx - dec_f32(m.y)));
            atomic_add_f32_hw(sb + 2, __expf( sy - dec_f32(m.z)));
            atomic_add_f32_hw(sb + 3, __expf(-sy - dec_f32(m.w)));
        }
    } else {
        for (int p = p0; p < P && p < p0 + 2; ++p) {
            float sx = pos[p] * inv_g;
            float sy = pos[P + p] * inv_g;
            int   n  = p2n[p];
            v4u m = *((const v4u*)enc_max + (unsigned)n);
            float* sb = sums + 4u * (unsigned)n;
            atomic_add_f32_hw(sb + 0, __expf( sx - dec_f32(m.x)));
            atomic_add_f32_hw(sb + 1, __expf(-sx - dec_f32(m.y)));
            atomic_add_f32_hw(sb + 2, __expf( sy - dec_f32(m.z)));
            atomic_add_f32_hw(sb + 3, __expf(-sy - dec_f32(m.w)));
        }
    }
}

// ---------------------------------------------------------------------------
// Kernel 4: per-net combine (b128 NT loads: dead after read) + block
// reduction + one f32 atomicAdd per block.
// ---------------------------------------------------------------------------
__global__ void k_finalize(const v4u* __restrict__ enc_max4,
                           const v4f* __restrict__ sums4,
                           const float* __restrict__ w,
                           const unsigned char* __restrict__ mask,
                           const float* __restrict__ gamma,
                           float* __restrict__ out,
                           int N) {
    __shared__ float red[THREADS];
    float acc = 0.0f;
    for (int n = blockIdx.x * blockDim.x + threadIdx.x; n < N;
         n += gridDim.x * blockDim.x) {
        v4u m = __builtin_nontemporal_load(enc_max4 + n);
        if (m.x != ENC_NEG_INF && mask[n]) {
            v4f s = __builtin_nontemporal_load(sums4 + n);
            float lse = (dec_f32(m.x) + logf(s.x)) + (dec_f32(m.y) + logf(s.y)) +
                        (dec_f32(m.z) + logf(s.z)) + (dec_f32(m.w) + logf(s.w));
            acc += w[n] * lse;
        }
    }
    red[threadIdx.x] = acc;
    __syncthreads();
    for (int s = THREADS / 2; s > 0; s >>= 1) {
        if (threadIdx.x < (unsigned)s) red[threadIdx.x] += red[threadIdx.x + s];
        __syncthreads();
    }
    if (threadIdx.x == 0) atomic_add_f32_hw(out, gamma[0] * red[0]);
}

// ---------------------------------------------------------------------------
extern "C" void kernel_launch(void* const* d_in, const int* in_sizes, int n_in,
                              void* d_out, int out_size, void* d_ws, size_t ws_size,
                              hipStream_t stream) {
    const float*         pos   = (const float*)d_in[0];
    const int*           p2n   = (const int*)d_in[1];
    const float*         w     = (const float*)d_in[2];
    const unsigned char* mask  = (const unsigned char*)d_in[3]; // jnp bool = 1B
    // d_in[4] = pin_mask: unused in forward
    const float*         gamma = (const float*)d_in[5];
    float*               out   = (float*)d_out;

    const int P = in_sizes[1];   // 10M pins
    const int N = in_sizes[2];   // 2M nets

    // Workspace layout: [4*N u32 encoded maxes][4*N f32 sums]  (~64 MB)
    unsigned* enc_max = (unsigned*)d_ws;
    float*    sums    = (float*)((char*)d_ws + (size_t)4 * N * sizeof(unsigned));

    const int pairs  = (P + 1) / 2;
    const int gInit  = (N + THREADS - 1) / THREADS;
    const int gPins  = (pairs + THREADS - 1) / THREADS;
    const int gNets  = (N + THREADS - 1) / THREADS;

    k_init    <<<gInit, THREADS, 0, stream>>>((v4u*)enc_max, (v4f*)sums, out, N);
    k_max     <<<gPins, THREADS, 0, stream>>>(pos, p2n, gamma, enc_max, P);
    k_sum     <<<gPins, THREADS, 0, stream>>>(pos, p2n, gamma, enc_max, sums, P);
    k_finalize<<<gNets, THREADS, 0, stream>>>((const v4u*)enc_max, (const v4f*)sums,
                                              w, mask, gamma, out, N);
}